// CoherentLoss_37684043055203
// MI455X (gfx1250) — compile-verified
//
#include <hip/hip_runtime.h>
#include <math.h>

// ---------------------------------------------------------------------------
// CoherentLoss on gfx1250 (MI455X).
// gt integral mapped to V_WMMA_F32_16X16X4_F32: one wave owns a tile of 16
// trajectories; K-loop over grid in chunks of 4. A = broadcast trapezoid
// weights (norm*w*psi, zero-padded), B = generated Gaussian values (re/im),
// C = gt. x and weights are pre-padded so the hot loop is branch-free.
// Unique-rank via bitmap + prefix popcount. Everything f32 (kernel is
// transcendental-bound; memory traffic ~1MB is irrelevant vs 23.3 TB/s).
// ---------------------------------------------------------------------------

typedef __attribute__((ext_vector_type(2))) float v2f;
typedef __attribute__((ext_vector_type(8))) float v8f;

#define NORMC 0.8932438417066635f   // (2/pi)^0.25, gamma_f = 1
#define QMINC (-8.0f)
#define PMINC (-10.0f)
#define DQC 0.125f                  // (QMAX-QMIN)/QBINS = 16/128
#define DPC 0.15625f                // (PMAX-PMIN)/PBINS = 20/128
#define PBINSC 128
#define BMAP_WORDS 32768            // 2^20 bits of bin presence
#define BIN_OFFSET (1 << 19)

// -------------------- kernel 0: zero scratch that must start at 0 ----------
__global__ void k_zero(unsigned* bitmap, float* bre, float* bim, float* lossacc,
                       int N) {
  int i = blockIdx.x * blockDim.x + threadIdx.x;
  if (i < BMAP_WORDS) bitmap[i] = 0u;
  if (i < N) { bre[i] = 0.f; bim[i] = 0.f; }
  if (i == 0) *lossacc = 0.f;
}

// ------ kernel 1: trapezoid weights * psi * norm (zero-padded) + padded x --
__global__ void k_wpsi(const float* __restrict__ x, const float* __restrict__ psi,
                       float* __restrict__ wpsi, float* __restrict__ xp,
                       int M, int Mpad) {
  int m = blockIdx.x * blockDim.x + threadIdx.x;
  if (m >= Mpad) return;
  if (m < M) {
    float w = 0.f;
    if (m + 1 < M) w += x[m + 1] - x[m];
    if (m > 0)     w += x[m] - x[m - 1];
    wpsi[m] = NORMC * 0.5f * w * psi[m];
    xp[m] = x[m];
  } else {
    wpsi[m] = 0.f;          // pad contributes nothing through A operand
    xp[m] = x[M - 1];       // finite value -> finite (but weightless) B
  }
}

// -------------------- kernel 2: per-trajectory prep ------------------------
__global__ void k_prep(const float* __restrict__ fre, const float* __restrict__ fim,
                       const float* __restrict__ qre, const float* __restrict__ qim,
                       const float* __restrict__ pre, const float* __restrict__ pim,
                       float* __restrict__ qc, float* __restrict__ pc,
                       float* __restrict__ vre, float* __restrict__ vim,
                       int* __restrict__ bins, int N) {
  int n = blockIdx.x * blockDim.x + threadIdx.x;
  if (n >= N) return;
  float qi = qim[n], pr = pre[n];
  float qf = qre[n] - 0.5f * pim[n];        // Re(xi)/(2g), g=1
  float pf = 2.f * qi + pr;                 // Im(xi)
  float qb = floorf((qf - QMINC) / DQC);
  float pb = floorf((pf - PMINC) / DPC);
  bins[n] = (int)(qb * (float)PBINSC + pb);
  qc[n] = (qb + 0.5f) * DQC + QMINC;
  pc[n] = (pb + 0.5f) * DPC + PMINC;
  // pref = norm * exp(g*qi^2 + i*pr*qi), nan_to_num then clip to [-100,100]
  float amp = NORMC * expf(qi * qi);
  float ph = pr * qi;
  float prr = amp * cosf(ph);
  float pri = amp * sinf(ph);
  prr = isnan(prr) ? 0.f : prr;
  pri = isnan(pri) ? 0.f : pri;
  prr = fminf(fmaxf(prr, -100.f), 100.f);
  pri = fminf(fmaxf(pri, -100.f), 100.f);
  // vals = pref * factors (complex)
  float fr = fre[n], fi = fim[n];
  vre[n] = prr * fr - pri * fi;
  vim[n] = prr * fi + pri * fr;
}

// -------------------- kernel 3: gt via WMMA f32 16x16x4 --------------------
// One wave per tile of 16 trajectories. Lane layout (32-bit A/B operands):
// lane l: row/col index j = l&15, half h = l>>4; reg r holds K = r + 2h.
// A rows are the broadcast weight vector -> D rows identical; lane<16 reads
// D VGPR0 (M=0, N=lane) at the end. Hot loop is branch-free: padded arrays,
// aligned v2f loads of the contiguous pair {m0+2h, m0+2h+1}.
__global__ void __launch_bounds__(256)
k_gt(const float* __restrict__ xp, const float* __restrict__ wpsi,
     const float* __restrict__ qc, const float* __restrict__ pc,
     float* __restrict__ gtre, float* __restrict__ gtim, int Mpad, int N) {
  const int wave  = threadIdx.x >> 5;
  const int lane  = threadIdx.x & 31;
  const int tiles = (N + 15) >> 4;
  const int tile  = blockIdx.x * (blockDim.x >> 5) + wave;
  if (tile >= tiles) return;                 // wave-uniform exit
  const int j  = lane & 15;
  const int kh = lane >> 4;
  const int jg = tile * 16 + j;
  const int jgc = (jg < N) ? jg : (N - 1);   // clamp for partial tiles
  const float qcj = qc[jgc];
  const float pcj = pc[jgc];

  const v2f* __restrict__ wp2 = (const v2f*)wpsi;
  const v2f* __restrict__ xp2 = (const v2f*)xp;

  v8f cre = {};
  v8f cim = {};
#pragma unroll 4
  for (int m0 = 0; m0 < Mpad; m0 += 4) {
    const int i2 = (m0 >> 1) + kh;           // pair {m0+2h, m0+2h+1}
    const v2f a  = wp2[i2];                  // A: K = 2h, 2h+1 (broadcast rows)
    const v2f xv = xp2[i2];
    const float d0 = xv.x - qcj, d1 = xv.y - qcj;
    const float e0 = __expf(-d0 * d0);
    const float e1 = __expf(-d1 * d1);
    float s0, c0, s1, c1;
    __sincosf(pcj * d0, &s0, &c0);
    __sincosf(pcj * d1, &s1, &c1);
    v2f br = {e0 * c0, e1 * c1};             // Re conj(gf)
    v2f bi = {-e0 * s0, -e1 * s1};           // Im conj(gf)
    cre = __builtin_amdgcn_wmma_f32_16x16x4_f32(false, a, false, br,
                                                (short)0, cre, false, false);
    cim = __builtin_amdgcn_wmma_f32_16x16x4_f32(false, a, false, bi,
                                                (short)0, cim, false, false);
  }
  if (lane < 16 && jg < N) {
    gtre[jg] = cre[0];                       // D row M=0, N=lane
    gtim[jg] = cim[0];
  }
}

// -------------------- kernel 4: mark bin presence bitmap -------------------
__global__ void k_mark(const int* __restrict__ bins, unsigned* __restrict__ bitmap,
                       int N) {
  int n = blockIdx.x * blockDim.x + threadIdx.x;
  if (n >= N) return;
  long long v = (long long)bins[n] + BIN_OFFSET;
  if (v < 0) v = 0;
  if (v > (long long)BMAP_WORDS * 32 - 1) v = (long long)BMAP_WORDS * 32 - 1;
  int idx = (int)v;
  atomicOr(&bitmap[idx >> 5], 1u << (idx & 31));
}

// -------------------- kernel 5: prefix popcount scan (one block) -----------
__global__ void k_scan(const unsigned* __restrict__ bitmap, int* __restrict__ wpref) {
  __shared__ int part[1024];
  const int t = threadIdx.x;
  const int WPT = BMAP_WORDS / 1024;        // 32 words per thread
  const int base = t * WPT;
  int s = 0;
  for (int i = 0; i < WPT; ++i) s += __popc(bitmap[base + i]);
  part[t] = s;
  __syncthreads();
  for (int off = 1; off < 1024; off <<= 1) {  // Hillis-Steele inclusive scan
    int v = (t >= off) ? part[t - off] : 0;
    __syncthreads();
    part[t] += v;
    __syncthreads();
  }
  int run = (t == 0) ? 0 : part[t - 1];       // exclusive prefix
  for (int i = 0; i < WPT; ++i) {
    wpref[base + i] = run;
    run += __popc(bitmap[base + i]);
  }
}

// -------------------- kernel 6: rank + scatter-add vals --------------------
__global__ void k_scatter(const int* __restrict__ bins,
                          const unsigned* __restrict__ bitmap,
                          const int* __restrict__ wpref,
                          const float* __restrict__ vre, const float* __restrict__ vim,
                          float* __restrict__ bre, float* __restrict__ bim, int N) {
  int n = blockIdx.x * blockDim.x + threadIdx.x;
  if (n >= N) return;
  long long v = (long long)bins[n] + BIN_OFFSET;
  if (v < 0) v = 0;
  if (v > (long long)BMAP_WORDS * 32 - 1) v = (long long)BMAP_WORDS * 32 - 1;
  int idx = (int)v;
  int w = idx >> 5, b = idx & 31;
  unsigned below = (b == 0) ? 0u : (0xFFFFFFFFu >> (32 - b));
  int rank = wpref[w] + __popc(bitmap[w] & below);
  atomicAdd(&bre[rank], vre[n]);
  atomicAdd(&bim[rank], vim[n]);
}

// -------------------- kernel 7: loss reduction -----------------------------
__global__ void k_loss(const float* __restrict__ bre, const float* __restrict__ bim,
                       const float* __restrict__ gtre, const float* __restrict__ gtim,
                       float* __restrict__ lossacc, int N) {
  __shared__ float red[256];
  int n = blockIdx.x * blockDim.x + threadIdx.x;
  float v = 0.f;
  if (n < N) {
    float dr = bre[n] - gtre[n];
    float di = bim[n] - gtim[n];
    v = dr * dr + di * di;
  }
  red[threadIdx.x] = v;
  __syncthreads();
  for (int off = 128; off > 0; off >>= 1) {
    if (threadIdx.x < (unsigned)off) red[threadIdx.x] += red[threadIdx.x + off];
    __syncthreads();
  }
  if (threadIdx.x == 0) atomicAdd(lossacc, red[0]);
}

__global__ void k_final(const float* __restrict__ lossacc, float* __restrict__ out) {
  if (threadIdx.x == 0 && blockIdx.x == 0) out[0] = sqrtf(*lossacc);
}

// ---------------------------------------------------------------------------
extern "C" void kernel_launch(void* const* d_in, const int* in_sizes, int n_in,
                              void* d_out, int out_size, void* d_ws, size_t ws_size,
                              hipStream_t stream) {
  const float* fre = (const float*)d_in[0];
  const float* fim = (const float*)d_in[1];
  const float* qre = (const float*)d_in[2];
  const float* qim = (const float*)d_in[3];
  const float* pre = (const float*)d_in[4];
  const float* pim = (const float*)d_in[5];
  const float* x   = (const float*)d_in[6];
  const float* psi = (const float*)d_in[7];
  float* out = (float*)d_out;

  const int N = in_sizes[0];     // 16384 trajectories
  const int M = in_sizes[6];     // ~2401 grid points
  const int Mpad = (M + 3) & ~3; // multiple of 4 for K-chunks

  // ---- workspace partition (floats/ints, all properly aligned) ----
  float* ws = (float*)d_ws;
  size_t off = 0;
  float* wpsi = ws + off; off += (size_t)Mpad;
  float* xp   = ws + off; off += (size_t)Mpad;
  float* qc   = ws + off; off += N;
  float* pc   = ws + off; off += N;
  float* vre  = ws + off; off += N;
  float* vim  = ws + off; off += N;
  int*   bins = (int*)(ws + off); off += N;
  unsigned* bitmap = (unsigned*)(ws + off); off += BMAP_WORDS;
  int*   wpref = (int*)(ws + off); off += BMAP_WORDS;
  float* gtre = ws + off; off += N;
  float* gtim = ws + off; off += N;
  float* bre  = ws + off; off += N;
  float* bim  = ws + off; off += N;
  float* lossacc = ws + off; off += 1;

  const int B = 256;
  int zmax = (N > BMAP_WORDS) ? N : BMAP_WORDS;
  k_zero<<<(zmax + B - 1) / B, B, 0, stream>>>(bitmap, bre, bim, lossacc, N);
  k_wpsi<<<(Mpad + B - 1) / B, B, 0, stream>>>(x, psi, wpsi, xp, M, Mpad);
  k_prep<<<(N + B - 1) / B, B, 0, stream>>>(fre, fim, qre, qim, pre, pim,
                                            qc, pc, vre, vim, bins, N);
  const int tiles = (N + 15) / 16;
  const int wavesPerBlk = B / 32;           // 8 tiles per 256-thread block
  k_gt<<<(tiles + wavesPerBlk - 1) / wavesPerBlk, B, 0, stream>>>(
      xp, wpsi, qc, pc, gtre, gtim, Mpad, N);
  k_mark<<<(N + B - 1) / B, B, 0, stream>>>(bins, bitmap, N);
  k_scan<<<1, 1024, 0, stream>>>(bitmap, wpref);
  k_scatter<<<(N + B - 1) / B, B, 0, stream>>>(bins, bitmap, wpref, vre, vim,
                                               bre, bim, N);
  k_loss<<<(N + B - 1) / B, B, 0, stream>>>(bre, bim, gtre, gtim, lossacc, N);
  k_final<<<1, 1, 0, stream>>>(lossacc, out);
}